// CNN_SeqCap_66391604461671
// MI455X (gfx1250) — compile-verified
//
#include <hip/hip_runtime.h>
#include <hip/hip_bf16.h>

typedef float v2f __attribute__((ext_vector_type(2)));
typedef float v8f __attribute__((ext_vector_type(8)));

// ---------------- geometry constants ----------------
#define BATCH 8
#define H_IN 80
#define T_IN 2044
#define H1 40
#define T1 1022          // after conv1(5x5,pad2)+pool2
#define C1 16
#define H2 20
#define T2 511           // after conv2(3x3,pad1)+pool2
#define C2 32
#define CP 256
#define HP 9
#define TP 255           // prim conv 3x3 stride2 pad0
#define NW 50            // windows per batch
#define NWIN 400         // 8*50
#define ICAP 2880        // input caps per window
#define PDIM 8
#define NCLS 4
#define ODIM 16

// ---------------- kernel 1: conv1 (5x5,pad2) + relu + maxpool2 ----------------
__global__ void conv1_pool_k(const float* __restrict__ x, const float* __restrict__ w1,
                             const float* __restrict__ b1, float* __restrict__ f1) {
  int idx = blockIdx.x * 256 + threadIdx.x;
  const int total = BATCH * C1 * H1 * T1;
  if (idx >= total) return;
  int xo = idx % T1; int t = idx / T1;
  int y = t % H1; t /= H1;
  int co = t % C1; int b = t / C1;
  const float* wr = w1 + co * 25;
  float bias = b1[co];
  const float* xin = x + (long)b * H_IN * T_IN;
  float best = -3.4e38f;
  for (int py = 0; py < 2; ++py)
    for (int px = 0; px < 2; ++px) {
      int ry = 2 * y + py, cx = 2 * xo + px;
      float acc = bias;
      for (int ky = 0; ky < 5; ++ky) {
        int r = ry + ky - 2;
        if ((unsigned)r >= H_IN) continue;
        const float* rowp = xin + r * T_IN;
        for (int kx = 0; kx < 5; ++kx) {
          int c = cx + kx - 2;
          if ((unsigned)c >= T_IN) continue;
          acc += wr[ky * 5 + kx] * rowp[c];
        }
      }
      acc = fmaxf(acc, 0.f);
      best = fmaxf(best, acc);
    }
  f1[idx] = best;
}

// ---------------- kernel 2: conv2 (3x3,pad1) + relu + maxpool2, LDS tiled ----------------
// grid (xtiles=8, ytiles=10, b=8), block 128 = 64x * 2y
__global__ void conv2_pool_k(const float* __restrict__ f1, const float* __restrict__ w2,
                             const float* __restrict__ b2, float* __restrict__ f2) {
  int b = blockIdx.z, y0 = blockIdx.y * 2, x0 = blockIdx.x * 64;
  int xl = threadIdx.x & 63, yl = threadIdx.x >> 6;
  __shared__ float tile[C1][6][130];   // 48.8 KB
  for (int e = threadIdx.x; e < C1 * 6 * 130; e += 128) {
    int c = e % 130; int rr = (e / 130) % 6; int ci = e / (130 * 6);
    int gr = 2 * y0 - 1 + rr;
    int gc = 2 * x0 - 1 + c;
    float v = 0.f;
    if ((unsigned)gr < H1 && (unsigned)gc < T1)
      v = f1[((b * C1 + ci) * H1 + gr) * T1 + gc];
    tile[ci][rr][c] = v;
  }
  __syncthreads();
  int xo = x0 + xl, yo = y0 + yl;
  bool valid = (xo < T2);
  for (int co = 0; co < C2; ++co) {
    float bias = b2[co];
    float a00 = bias, a01 = bias, a10 = bias, a11 = bias;
    const float* w = w2 + co * C1 * 9;
    for (int ci = 0; ci < C1; ++ci)
      for (int ky = 0; ky < 3; ++ky) {
        const float* trow0 = &tile[ci][2 * yl + ky][2 * xl];
        const float* trow1 = &tile[ci][2 * yl + 1 + ky][2 * xl];
        for (int kx = 0; kx < 3; ++kx) {
          float wv = w[(ci * 3 + ky) * 3 + kx];
          a00 += wv * trow0[kx];
          a01 += wv * trow0[1 + kx];
          a10 += wv * trow1[kx];
          a11 += wv * trow1[1 + kx];
        }
      }
    float r = fmaxf(fmaxf(fmaxf(a00, 0.f), fmaxf(a01, 0.f)),
                    fmaxf(fmaxf(a10, 0.f), fmaxf(a11, 0.f)));
    if (valid) f2[((b * C2 + co) * H2 + yo) * T2 + xo] = r;
  }
}

// ---------------- kernel 3: prim-caps conv as implicit GEMM with WMMA ----------------
// out[co, (b,h,t)] = sum_{k=(ci,kh,kw)} wp[co,k] * f2[b,ci,2h+kh,2t+kw]
// grid (co_tiles=16, h=9, b=8), block 256 (8 waves); each wave: 2 t-tiles of 16
__global__ void convp_wmma_k(const float* __restrict__ f2, const float* __restrict__ wp,
                             const float* __restrict__ bp, float* __restrict__ pc) {
  int co0 = blockIdx.x * 16;
  int h = blockIdx.y, b = blockIdx.z;
  int lane = threadIdx.x & 31, wave = threadIdx.x >> 5;
  int row = lane & 15, half = lane >> 4;
  __shared__ float ALds[288][16];      // 18.4 KB: ALds[k][co_local]
  for (int e = threadIdx.x; e < 288 * 16; e += 256) {
    int k = e >> 4, c = e & 15;
    ALds[k][c] = wp[(co0 + c) * 288 + k];
  }
  __syncthreads();
  const float* f2b = f2 + (long)b * C2 * H2 * T2;
  for (int tt = 0; tt < 2; ++tt) {
    int t0 = (wave * 2 + tt) * 16;
    int t = t0 + row;
    int tl = t > (TP - 1) ? (TP - 1) : t;   // clamp loads for t==255
    v8f C = {};
    for (int kg = 0; kg < 72; ++kg) {
      int kb = kg * 4 + 2 * half;
      v2f A, B;
      A.x = ALds[kb][row];
      A.y = ALds[kb + 1][row];
      {
        int k = kb;
        int ci = k / 9, r9 = k % 9, kh = r9 / 3, kw = r9 % 3;
        B.x = f2b[(ci * H2 + (2 * h + kh)) * T2 + 2 * tl + kw];
        k = kb + 1;
        ci = k / 9; r9 = k % 9; kh = r9 / 3; kw = r9 % 3;
        B.y = f2b[(ci * H2 + (2 * h + kh)) * T2 + 2 * tl + kw];
      }
      C = __builtin_amdgcn_wmma_f32_16x16x4_f32(false, A, false, B, (short)0, C, false, false);
    }
    if (t < TP) {
      #pragma unroll
      for (int r = 0; r < 8; ++r) {
        int co = co0 + r + 8 * half;
        pc[((b * CP + co) * HP + h) * TP + t] = C[r] + bp[co];
      }
    }
  }
}

// ---------------- kernel 4: window gather -> uT[cap][k][w] (w fastest, 400) ----------------
__global__ void win_gather_k(const float* __restrict__ pc, float* __restrict__ uT) {
  int idx = blockIdx.x * 256 + threadIdx.x;
  const int total = ICAP * PDIM * NWIN;     // 9,216,000
  if (idx >= total) return;
  int w = idx % NWIN;
  int j = idx / NWIN;            // cap*8+k == flat (ch,tt) index
  int ch = j / 10, ttt = j % 10;
  int c = ch / HP, hh = ch % HP;
  int b = w / NW, win = w % NW;
  int t = win * 5 + ttt;
  uT[idx] = pc[((b * CP + c) * HP + hh) * TP + t];
}

// ---------------- kernel 5: zero ----------------
__global__ void zero_k(float* __restrict__ p, int n) {
  int i = blockIdx.x * 256 + threadIdx.x;
  if (i < n) p[i] = 0.f;
}

// ---------------- kernel 6: routing S pass (windows batched as WMMA M dim) ----------------
// s[w,o,d] = sum_i softmax_o(b[w,i,:]) * (sum_k Ww[i,o,d,k]*u[w,i,k])
// grid (wt=25, o=4), block 256 (8 waves); A = c*u, accumulate WMMA over i
__global__ void route_s_k(const float* __restrict__ uT, const float* __restrict__ Ww,
                          const float* __restrict__ blog, float* __restrict__ sbuf) {
  int wt = blockIdx.x, o = blockIdx.y;
  int w0 = wt * 16;
  int lane = threadIdx.x & 31, wave = threadIdx.x >> 5;
  int row = lane & 15, half = lane >> 4;
  int widx = w0 + row;
  const float* bptr = blog + (long)widx * ICAP * NCLS;
  v8f C = {};
  for (int i = wave; i < ICAP; i += 8) {
    float b0 = bptr[i * 4 + 0], b1 = bptr[i * 4 + 1];
    float b2 = bptr[i * 4 + 2], b3 = bptr[i * 4 + 3];
    float m = fmaxf(fmaxf(b0, b1), fmaxf(b2, b3));
    float e0 = __expf(b0 - m), e1 = __expf(b1 - m);
    float e2 = __expf(b2 - m), e3 = __expf(b3 - m);
    float eo = (o == 0) ? e0 : (o == 1) ? e1 : (o == 2) ? e2 : e3;
    float c = eo / (e0 + e1 + e2 + e3);
    const float* up = uT + ((long)i * PDIM + 2 * half) * NWIN + widx;
    v2f A0, A1, B0, B1;
    A0.x = up[0] * c;            A0.y = up[NWIN] * c;
    A1.x = up[4 * NWIN] * c;     A1.y = up[5 * NWIN] * c;
    const float* wpr = Ww + (((long)i * NCLS + o) * ODIM + row) * PDIM + 2 * half;
    B0.x = wpr[0]; B0.y = wpr[1];
    B1.x = wpr[4]; B1.y = wpr[5];
    C = __builtin_amdgcn_wmma_f32_16x16x4_f32(false, A0, false, B0, (short)0, C, false, false);
    C = __builtin_amdgcn_wmma_f32_16x16x4_f32(false, A1, false, B1, (short)0, C, false, false);
  }
  __shared__ float red[8][32][8];
  #pragma unroll
  for (int r = 0; r < 8; ++r) red[wave][lane][r] = C[r];
  __syncthreads();
  int t = threadIdx.x;
  int lane_ = t >> 3, r_ = t & 7;
  float s = 0.f;
  #pragma unroll
  for (int wv = 0; wv < 8; ++wv) s += red[wv][lane_][r_];
  int rowM = r_ + 8 * (lane_ >> 4);
  int d = lane_ & 15;
  sbuf[((w0 + rowM) * NCLS + o) * ODIM + d] = s;
}

// ---------------- kernel 7: squash ----------------
__global__ void squash_k(const float* __restrict__ sbuf, float* __restrict__ vbuf) {
  int idx = blockIdx.x * 256 + threadIdx.x;
  if (idx >= NWIN * NCLS) return;
  const float* s = sbuf + idx * ODIM;
  float sn = 0.f;
  #pragma unroll
  for (int d = 0; d < ODIM; ++d) sn += s[d] * s[d];
  float sc = sn / ((1.f + sn) * (sqrtf(sn) + 1e-8f));
  float* v = vbuf + idx * ODIM;
  #pragma unroll
  for (int d = 0; d < ODIM; ++d) v[d] = s[d] * sc;
}

// ---------------- kernel 8: routing B pass: b[w,i,o] += u_hat[w,i,o,:] . v[w,o,:] ----------------
__global__ void route_b_k(const float* __restrict__ uT, const float* __restrict__ Ww,
                          const float* __restrict__ vbuf, float* __restrict__ blog) {
  int wt = blockIdx.x, o = blockIdx.y;
  int w0 = wt * 16;
  int lane = threadIdx.x & 31, wave = threadIdx.x >> 5;
  int row = lane & 15, half = lane >> 4;
  int widx = w0 + row;
  __shared__ float vt[16][16];
  {
    int j = threadIdx.x >> 4, d = threadIdx.x & 15;
    vt[j][d] = vbuf[((w0 + j) * NCLS + o) * ODIM + d];
  }
  __syncthreads();
  for (int i = wave; i < ICAP; i += 8) {
    const float* up = uT + ((long)i * PDIM + 2 * half) * NWIN + widx;
    v2f A0, A1, B0, B1;
    A0.x = up[0];            A0.y = up[NWIN];
    A1.x = up[4 * NWIN];     A1.y = up[5 * NWIN];
    const float* wpr = Ww + (((long)i * NCLS + o) * ODIM + row) * PDIM + 2 * half;
    B0.x = wpr[0]; B0.y = wpr[1];
    B1.x = wpr[4]; B1.y = wpr[5];
    v8f C = {};
    C = __builtin_amdgcn_wmma_f32_16x16x4_f32(false, A0, false, B0, (short)0, C, false, false);
    C = __builtin_amdgcn_wmma_f32_16x16x4_f32(false, A1, false, B1, (short)0, C, false, false);
    #pragma unroll
    for (int r = 0; r < 8; ++r) {
      int rowM = r + 8 * half;
      float tval = C[r] * vt[rowM][row];   // lane&15 is d in C/D layout
      tval += __shfl_xor(tval, 1);
      tval += __shfl_xor(tval, 2);
      tval += __shfl_xor(tval, 4);
      tval += __shfl_xor(tval, 8);
      if (row == 0) {
        float* bp_ = blog + ((long)(w0 + rowM) * ICAP + i) * NCLS + o;
        *bp_ += tval;
      }
    }
  }
}

// ---------------- kernel 9: utterance input = mean over classes ----------------
__global__ void utt_mean_k(const float* __restrict__ vbuf, float* __restrict__ uttU) {
  int idx = blockIdx.x * 256 + threadIdx.x;   // 8*50*16
  if (idx >= BATCH * NW * ODIM) return;
  int d = idx & 15;
  int bi = idx >> 4;
  const float* v = vbuf + (long)bi * NCLS * ODIM;
  uttU[idx] = 0.25f * (v[d] + v[ODIM + d] + v[2 * ODIM + d] + v[3 * ODIM + d]);
}

// ---------------- kernel 10: utterance-level routing (tiny, LDS resident) ----------------
__global__ void utt_routing_k(const float* __restrict__ uttU, const float* __restrict__ Wu,
                              float* __restrict__ out) {
  int b = blockIdx.x;
  int tid = threadIdx.x;   // 256
  __shared__ float u[NW * 16];
  __shared__ float uhat[NW * NCLS * ODIM];
  __shared__ float blds[NW * NCLS];
  __shared__ float cl[NW * NCLS];
  __shared__ float sv[NCLS * ODIM];
  __shared__ float scal[NCLS];
  for (int e = tid; e < NW * 16; e += 256) u[e] = uttU[b * NW * 16 + e];
  for (int e = tid; e < NW * NCLS; e += 256) blds[e] = 0.f;
  __syncthreads();
  for (int e = tid; e < NW * NCLS * ODIM; e += 256) {
    int i = e >> 6; int o = (e >> 4) & 3; int d = e & 15;
    const float* wrow = Wu + ((i * NCLS + o) * ODIM + d) * 16;
    float acc = 0.f;
    #pragma unroll
    for (int k = 0; k < 16; ++k) acc += wrow[k] * u[i * 16 + k];
    uhat[e] = acc;
  }
  __syncthreads();
  for (int it = 0; it < 3; ++it) {
    for (int i = tid; i < NW; i += 256) {
      float b0 = blds[i * 4], b1 = blds[i * 4 + 1];
      float b2 = blds[i * 4 + 2], b3 = blds[i * 4 + 3];
      float m = fmaxf(fmaxf(b0, b1), fmaxf(b2, b3));
      float e0 = __expf(b0 - m), e1 = __expf(b1 - m);
      float e2 = __expf(b2 - m), e3 = __expf(b3 - m);
      float inv = 1.f / (e0 + e1 + e2 + e3);
      cl[i * 4] = e0 * inv; cl[i * 4 + 1] = e1 * inv;
      cl[i * 4 + 2] = e2 * inv; cl[i * 4 + 3] = e3 * inv;
    }
    __syncthreads();
    if (tid < 64) {
      int o = tid >> 4, d = tid & 15;
      float s = 0.f;
      for (int i = 0; i < NW; ++i) s += cl[i * 4 + o] * uhat[(i * 4 + o) * ODIM + d];
      sv[tid] = s;
    }
    __syncthreads();
    if (tid < 4) {
      float sn = 0.f;
      for (int d = 0; d < ODIM; ++d) { float xx = sv[tid * ODIM + d]; sn += xx * xx; }
      scal[tid] = sn / ((1.f + sn) * (sqrtf(sn) + 1e-8f));
    }
    __syncthreads();
    if (tid < 64) sv[tid] *= scal[tid >> 4];
    __syncthreads();
    if (it < 2) {
      for (int e = tid; e < NW * NCLS; e += 256) {
        int i = e >> 2, o = e & 3;
        float acc = 0.f;
        #pragma unroll
        for (int d = 0; d < ODIM; ++d) acc += uhat[(i * 4 + o) * ODIM + d] * sv[o * ODIM + d];
        blds[e] += acc;
      }
      __syncthreads();
    }
  }
  if (tid < 4) {
    float sn = 0.f;
    for (int d = 0; d < ODIM; ++d) { float xx = sv[tid * ODIM + d]; sn += xx * xx; }
    out[b * 4 + tid] = sqrtf(sn);
  }
}

// ---------------- host launch ----------------
extern "C" void kernel_launch(void* const* d_in, const int* in_sizes, int n_in,
                              void* d_out, int out_size, void* d_ws, size_t ws_size,
                              hipStream_t stream) {
  const float* x  = (const float*)d_in[0];
  const float* w1 = (const float*)d_in[1];
  const float* b1 = (const float*)d_in[2];
  const float* w2 = (const float*)d_in[3];
  const float* b2 = (const float*)d_in[4];
  const float* wp = (const float*)d_in[5];
  const float* bp = (const float*)d_in[6];
  const float* Ww = (const float*)d_in[7];
  const float* Wu = (const float*)d_in[8];
  float* out = (float*)d_out;

  float* ws = (float*)d_ws;
  const size_t SZ_F1 = (size_t)BATCH * C1 * H1 * T1;       // 5,232,640
  const size_t SZ_F2 = (size_t)BATCH * C2 * H2 * T2;       // 2,616,320
  const size_t SZ_PC = (size_t)BATCH * CP * HP * TP;       // 4,700,160
  const size_t SZ_UT = (size_t)ICAP * PDIM * NWIN;         // 9,216,000
  const size_t SZ_BL = (size_t)NWIN * ICAP * NCLS;         // 4,608,000
  const size_t SZ_SV = (size_t)NWIN * NCLS * ODIM;         // 25,600
  float* f1   = ws;                 ws += SZ_F1;
  float* f2   = ws;                 ws += SZ_F2;
  float* pc   = ws;                 ws += SZ_PC;
  float* uT   = ws;                 ws += SZ_UT;
  float* blog = ws;                 ws += SZ_BL;
  float* sbuf = ws;                 ws += SZ_SV;
  float* vbuf = ws;                 ws += SZ_SV;
  float* uttU = ws;                 ws += (size_t)BATCH * NW * ODIM;

  // 1) conv1 + relu + pool
  {
    int total = BATCH * C1 * H1 * T1;
    conv1_pool_k<<<(total + 255) / 256, 256, 0, stream>>>(x, w1, b1, f1);
  }
  // 2) conv2 + relu + pool (LDS tiled)
  conv2_pool_k<<<dim3(8, 10, BATCH), 128, 0, stream>>>(f1, w2, b2, f2);
  // 3) prim caps conv via WMMA implicit GEMM
  convp_wmma_k<<<dim3(CP / 16, HP, BATCH), 256, 0, stream>>>(f2, wp, bp, pc);
  // 4) windowing gather into uT[cap][k][w]
  {
    int total = ICAP * PDIM * NWIN;
    win_gather_k<<<(total + 255) / 256, 256, 0, stream>>>(pc, uT);
  }
  // 5) zero routing logits (must be re-zeroed every call)
  zero_k<<<((int)SZ_BL + 255) / 256, 256, 0, stream>>>(blog, (int)SZ_BL);
  // 6) window-level dynamic routing: 3 iters (skip last b-update, unused)
  for (int it = 0; it < 3; ++it) {
    route_s_k<<<dim3(NWIN / 16, NCLS), 256, 0, stream>>>(uT, Ww, blog, sbuf);
    squash_k<<<(NWIN * NCLS + 255) / 256, 256, 0, stream>>>(sbuf, vbuf);
    if (it < 2)
      route_b_k<<<dim3(NWIN / 16, NCLS), 256, 0, stream>>>(uT, Ww, vbuf, blog);
  }
  // 7) utterance inputs = mean over classes
  utt_mean_k<<<(BATCH * NW * ODIM + 255) / 256, 256, 0, stream>>>(vbuf, uttU);
  // 8) utterance routing + output norms
  utt_routing_k<<<BATCH, 256, 0, stream>>>(uttU, Wu, out);

  (void)in_sizes; (void)n_in; (void)out_size; (void)ws_size;
}